// Center_of_mass_initial_pairwise_37263136260923
// MI455X (gfx1250) — compile-verified
//
#include <hip/hip_runtime.h>
#include <stdint.h>

#define N      192
#define N2     (N * N)
#define N3     (N * N * N)
#define BATCH  2
#define TOT    (BATCH * N3)            // elements per input tensor
#define TOT4   (TOT / 4)               // float4 quads per input tensor
#define GRID_OFF ((size_t)BATCH * N3)  // transformed size; grid starts here in d_out
#define GSTEP  (2.0f / 191.0f)

#define RED_BLOCKS  512                // 512*256 threads * 27 iters * 4 = TOT exactly
#define RED_THREADS 256
#define CONST_OFF   (8 * RED_BLOCKS)   // float index in d_ws where constants live

typedef __attribute__((ext_vector_type(2))) float v2f;
typedef __attribute__((ext_vector_type(8))) float v8f;

// -------- CDNA5 async global->LDS helpers (guarded; fallback = sync copy) ----
#if defined(__gfx1250__) && __has_builtin(__builtin_amdgcn_global_load_async_to_lds_b32)
#define HAS_ASYNC_LDS 1
typedef __attribute__((address_space(1))) int GInt;
typedef __attribute__((address_space(3))) int LInt;
__device__ __forceinline__ void async_copy_f32(float* lds_dst, const float* gsrc) {
  __builtin_amdgcn_global_load_async_to_lds_b32(
      (GInt*)(void*)gsrc, (LInt*)lds_dst, /*offset=*/0, /*cpol=*/0);
}
#else
#define HAS_ASYNC_LDS 0
__device__ __forceinline__ void async_copy_f32(float* lds_dst, const float* gsrc) {
  *lds_dst = *gsrc;
}
#endif

__device__ __forceinline__ void async_wait_all() {
#if HAS_ASYNC_LDS
#if __has_builtin(__builtin_amdgcn_s_wait_asynccnt)
  __builtin_amdgcn_s_wait_asynccnt(0);
#else
  asm volatile("s_wait_asynccnt 0" ::: "memory");
#endif
#endif
}

// -------- CDNA5 WMMA wave-sum (f32 16x16x4, "ones" row-sum trick) -----------
// A carries the per-lane value in both of its VGPRs; B = ones. Then
// D[m,n] = rowsum(A[m,:]) for every n. Per-lane sum of the 8 D VGPRs gives
// sum over one M-half at a fixed N; adding lane^16 gives sum over all M
// = sum of ALL 64 A elements = 2 * wave_sum(v). Robust to exact (m,k) layout.
// Requires EXEC == all ones (call only from fully-converged code).
#if defined(__gfx1250__) && __has_builtin(__builtin_amdgcn_wmma_f32_16x16x4_f32)
__device__ __forceinline__ float wave_sum(float v) {
  v2f a; a.x = v;    a.y = v;
  v2f b; b.x = 1.0f; b.y = 1.0f;
  v8f c = {};
  v8f d = __builtin_amdgcn_wmma_f32_16x16x4_f32(
      /*neg_a=*/false, a, /*neg_b=*/false, b,
      /*c_mod=*/(short)0, c, /*reuse_a=*/false, /*reuse_b=*/false);
  float t = ((d[0] + d[1]) + (d[2] + d[3])) + ((d[4] + d[5]) + (d[6] + d[7]));
  t += __shfl_xor(t, 16, 32);
  return 0.5f * t;
}
#else
__device__ __forceinline__ float wave_sum(float v) {
  #pragma unroll
  for (int off = 16; off > 0; off >>= 1) v += __shfl_down(v, off, 32);
  return v;   // valid in lane 0
}
#endif

// ---------------------------------------------------------------------------
// Kernel 1: 8 global weighted sums (deterministic: fixed per-block partials).
// float4-vectorized: a quad never straddles a row (192 % 4 == 0); gy/gz are
// constant within a quad and gx is affine in the intra-quad position.
// partials layout: p[c * RED_BLOCKS + blockIdx], c = 0..7:
//   0:Σx 1:Σx·gx 2:Σx·gy 3:Σx·gz 4:Σy 5:Σy·gx 6:Σy·gy 7:Σy·gz
// ---------------------------------------------------------------------------
__global__ __launch_bounds__(RED_THREADS) void reduce_kernel(
    const float4* __restrict__ x4, const float4* __restrict__ y4,
    float* __restrict__ partials) {
  float a0 = 0.f, a1 = 0.f, a2 = 0.f, a3 = 0.f;
  float b0 = 0.f, b1 = 0.f, b2 = 0.f, b3 = 0.f;
  const unsigned stride = gridDim.x * blockDim.x;        // 131072 quads
  for (unsigned q = blockIdx.x * blockDim.x + threadIdx.x; q < (unsigned)TOT4; q += stride) {
    __builtin_prefetch(&x4[q + stride], 0, 0);   // -> global_prefetch_b8 (speculative)
    __builtin_prefetch(&y4[q + stride], 0, 0);
    float4 xv = x4[q];
    float4 yv = y4[q];
    unsigned qr  = q % 48u;              // quad within row; d = qr*4
    unsigned row = q / 48u;
    unsigned w   = row % 192u;
    unsigned h   = (row / 192u) % 192u;  // batch index irrelevant for weights
    float gx0 = fmaf((float)(qr * 4u), GSTEP, -1.0f);
    float gy  = fmaf((float)w, GSTEP, -1.0f);
    float gz  = fmaf((float)h, GSTEP, -1.0f);
    float xs = ((xv.x + xv.y) + (xv.z + xv.w));
    float xp = fmaf(xv.w, 3.0f, fmaf(xv.z, 2.0f, xv.y));   // Σ k·x_k
    float ys = ((yv.x + yv.y) + (yv.z + yv.w));
    float yp = fmaf(yv.w, 3.0f, fmaf(yv.z, 2.0f, yv.y));
    a0 += xs;
    a1 = fmaf(xs, gx0, fmaf(xp, GSTEP, a1));
    a2 = fmaf(xs, gy, a2);
    a3 = fmaf(xs, gz, a3);
    b0 += ys;
    b1 = fmaf(ys, gx0, fmaf(yp, GSTEP, b1));
    b2 = fmaf(ys, gy, b2);
    b3 = fmaf(ys, gz, b3);
  }
  // Per-wave reduction (WMMA ones-trick; EXEC is all-ones here: uniform trip count)
  a0 = wave_sum(a0); a1 = wave_sum(a1); a2 = wave_sum(a2); a3 = wave_sum(a3);
  b0 = wave_sum(b0); b1 = wave_sum(b1); b2 = wave_sum(b2); b3 = wave_sum(b3);

  __shared__ float red[8][RED_THREADS / 32];   // [component][wave]
  const int wave = threadIdx.x >> 5;
  const int lane = threadIdx.x & 31;
  if (lane == 0) {
    red[0][wave] = a0; red[1][wave] = a1; red[2][wave] = a2; red[3][wave] = a3;
    red[4][wave] = b0; red[5][wave] = b1; red[6][wave] = b2; red[7][wave] = b3;
  }
  __syncthreads();
  if (threadIdx.x < 8) {
    float s = 0.f;
    #pragma unroll
    for (int wv = 0; wv < RED_THREADS / 32; ++wv) s += red[threadIdx.x][wv];
    partials[threadIdx.x * RED_BLOCKS + blockIdx.x] = s;
  }
}

// ---------------------------------------------------------------------------
// Kernel 2: deterministic final reduction + constant computation.
// cst (floats): 0:tx 1:ty 2:tz 3:fz 4:fy 5:fx 6:oz(int bits) 7:oy 8:ox
// ---------------------------------------------------------------------------
__global__ __launch_bounds__(256) void finalize_kernel(
    const float* __restrict__ partials, float* __restrict__ cst) {
  const int wave = threadIdx.x >> 5;   // component 0..7
  const int lane = threadIdx.x & 31;
  float s = 0.f;
  for (int i = lane; i < RED_BLOCKS; i += 32) s += partials[wave * RED_BLOCKS + i];
  s = wave_sum(s);
  __shared__ float comp[8];
  if (lane == 0) comp[wave] = s;
  __syncthreads();
  if (threadIdx.x == 0) {
    float xs = comp[0], xgx = comp[1], xgy = comp[2], xgz = comp[3];
    float ys = comp[4], ygx = comp[5], ygy = comp[6], ygz = comp[7];
    float tx = xgx / xs - ygx / ys;
    float ty = xgy / xs - ygy / ys;
    float tz = xgz / xs - ygz / ys;
    const float half = 0.5f * (float)(N - 1);      // 95.5
    float cx = tx * half, cy = ty * half, cz = tz * half;
    float fox = floorf(cx), foy = floorf(cy), foz = floorf(cz);
    cst[0] = tx; cst[1] = ty; cst[2] = tz;
    cst[3] = cz - foz;   // fz (outer / h axis, driven by tz)
    cst[4] = cy - foy;   // fy (middle / w axis, driven by ty)
    cst[5] = cx - fox;   // fx (inner / d axis, driven by tx)
    cst[6] = __int_as_float((int)foz);
    cst[7] = __int_as_float((int)foy);
    cst[8] = __int_as_float((int)fox);
  }
}

// ---------------------------------------------------------------------------
// Kernel 3: constant-shift trilinear resample + grid emission.
// One block per (b, z, y) output row; 192 threads = one x each.
// Stages the 4 clamped neighbor rows into LDS with async global->LDS loads.
// ---------------------------------------------------------------------------
__global__ __launch_bounds__(N) void sample_kernel(
    const float* __restrict__ vol, const float* __restrict__ cst,
    float* __restrict__ out) {
  __shared__ float rows[4 * N];

  const unsigned bid = blockIdx.x;
  const unsigned yq = bid % (unsigned)N;          // w index
  const unsigned t2 = bid / (unsigned)N;
  const unsigned zq = t2 % (unsigned)N;           // h index
  const unsigned bq = t2 / (unsigned)N;           // batch
  const int tid = threadIdx.x;                    // x (d axis)

  const float tx = cst[0], ty = cst[1], tz = cst[2];
  const float fz = cst[3], fy = cst[4], fx = cst[5];
  const int oz = __float_as_int(cst[6]);
  const int oy = __float_as_int(cst[7]);
  const int ox = __float_as_int(cst[8]);

  const float wz[2] = {1.0f - fz, fz};
  const float wy[2] = {1.0f - fy, fy};
  const float wx[2] = {1.0f - fx, fx};

  const float* base = vol + (size_t)bq * N3;
  const int z0 = (int)zq + oz;
  const int y0 = (int)yq + oy;

  float wrow[4];
  #pragma unroll
  for (int dz = 0; dz < 2; ++dz) {
    #pragma unroll
    for (int dy = 0; dy < 2; ++dy) {
      const int r = dz * 2 + dy;
      const int zc = z0 + dz, yc = y0 + dy;
      const bool valid = (zc >= 0) && (zc < N) && (yc >= 0) && (yc < N);
      wrow[r] = valid ? wz[dz] * wy[dy] : 0.0f;
      const int zcl = min(max(zc, 0), N - 1);
      const int ycl = min(max(yc, 0), N - 1);
      const float* rp = base + (size_t)zcl * N2 + (size_t)ycl * N;
      async_copy_f32(&rows[r * N + tid], rp + tid);   // per-lane async -> LDS
    }
  }
  async_wait_all();     // s_wait_asynccnt 0 (ASYNCcnt path)
  __syncthreads();

  float acc = 0.0f;
  #pragma unroll
  for (int r = 0; r < 4; ++r) {
    const float wr = wrow[r];
    #pragma unroll
    for (int dx = 0; dx < 2; ++dx) {
      const int xc = tid + ox + dx;
      if (xc >= 0 && xc < N) acc = fmaf(wr * wx[dx], rows[r * N + xc], acc);
    }
  }
  out[(size_t)bid * N + tid] = acc;

  // grid output: (b, h, w, d, 3); this row contributes 3*N contiguous floats.
  const size_t gbase = GRID_OFF + (size_t)bid * (3 * N);
  const float gy_v = fmaf((float)yq, GSTEP, -1.0f) + ty;
  const float gz_v = fmaf((float)zq, GSTEP, -1.0f) + tz;
  #pragma unroll
  for (int k = 0; k < 3; ++k) {
    const int j = tid + k * N;          // j in [0, 3N)
    const int pos = j / 3;
    const int comp = j - pos * 3;
    float g;
    if (comp == 0)      g = fmaf((float)pos, GSTEP, -1.0f) + tx;
    else if (comp == 1) g = gy_v;
    else                g = gz_v;
    out[gbase + j] = g;
  }
}

// ---------------------------------------------------------------------------
extern "C" void kernel_launch(void* const* d_in, const int* in_sizes, int n_in,
                              void* d_out, int out_size, void* d_ws, size_t ws_size,
                              hipStream_t stream) {
  const float* x = (const float*)d_in[0];
  const float* y = (const float*)d_in[1];
  float* out      = (float*)d_out;
  float* ws       = (float*)d_ws;
  float* partials = ws;               // 8 * RED_BLOCKS floats
  float* cst      = ws + CONST_OFF;   // 9 floats

  reduce_kernel<<<RED_BLOCKS, RED_THREADS, 0, stream>>>(
      (const float4*)x, (const float4*)y, partials);
  finalize_kernel<<<1, 256, 0, stream>>>(partials, cst);
  sample_kernel<<<BATCH * N2, N, 0, stream>>>(x, cst, out);
}